// LidarPillarTokenizer_39831526703842
// MI455X (gfx1250) — compile-verified
//
#include <hip/hip_runtime.h>
#include <math.h>

// ---- problem constants (match reference) ----
#define GX0   (-50.0f)
#define GY0   (-50.0f)
#define GCELL (0.5f)
#define GW    200
#define GH    200
#define KPIL  40000            // GW*GH
#define BATCH 16
#define NPTS  200000
#define BK    (BATCH * KPIL)   // 640000 pillars total
#define HID   64
#define EMB   96
#define LNEPS 1e-5f

#define NT 6                   // 96 / 16 N-tiles
#define KS 16                  // 64 / 4  K-steps

typedef __attribute__((ext_vector_type(2))) float v2f;
typedef __attribute__((ext_vector_type(8))) float v8f;

// --------------------------------------------------------------------------
// Kernel 1: zero the scatter accumulator (5 planes of BK floats in d_ws)
// --------------------------------------------------------------------------
__global__ void pillar_zero(float* __restrict__ acc, int n) {
  int i = blockIdx.x * blockDim.x + threadIdx.x;
  int stride = gridDim.x * blockDim.x;
  for (; i < n; i += stride) acc[i] = 0.0f;
}

// --------------------------------------------------------------------------
// Kernel 2: scatter points -> per-pillar {sumx,sumy,sumz,sumw,count}
// Accumulator layout: plane f at acc + f*BK (SoA, 12.8 MB -> L2-resident)
// --------------------------------------------------------------------------
__global__ void pillar_scatter(const float4* __restrict__ pts,
                               float* __restrict__ acc, int total) {
  int i = blockIdx.x * blockDim.x + threadIdx.x;
  if (i >= total) return;
  float4 p = pts[i];
  int xi = (int)floorf((p.x - GX0) * (1.0f / GCELL));
  int yi = (int)floorf((p.y - GY0) * (1.0f / GCELL));
  if (xi < 0 || xi >= GW || yi < 0 || yi >= GH) return;  // == ref's zero-scatter
  int b = i / NPTS;
  int g = b * KPIL + yi * GW + xi;
  unsafeAtomicAdd(&acc[0 * BK + g], p.x);   // global_atomic_add_f32 (no return)
  unsafeAtomicAdd(&acc[1 * BK + g], p.y);
  unsafeAtomicAdd(&acc[2 * BK + g], p.z);
  unsafeAtomicAdd(&acc[3 * BK + g], p.w);
  unsafeAtomicAdd(&acc[4 * BK + g], 1.0f);
}

// --------------------------------------------------------------------------
// Kernel 3: per-pillar MLP (5->64 relu ->96) + LayerNorm, fp32 WMMA path.
// One wave32 handles a tile of 16 pillars. Rolled K-loop keeps VGPR
// pressure low (no spills); 6 independent WMMAs per K-step fill the pipe.
// --------------------------------------------------------------------------
__global__ void __launch_bounds__(256)
pillar_mlp_ln(const float* __restrict__ acc,
              const float* __restrict__ w1,   // [5][64]
              const float* __restrict__ b1,   // [64]
              const float* __restrict__ w2,   // [64][96]
              const float* __restrict__ b2,   // [96]
              const float* __restrict__ gamma,// [96]
              const float* __restrict__ beta, // [96]
              float* __restrict__ out) {      // [BK][96]
  // w2 pre-swizzled into B-operand layout: [nt][s][lane] -> (b0,b1) pair
  __shared__ float s_w2[NT * KS * 32 * 2];    // 24 KB
  // w1/b1 packed per (k-step, lane-half): 12 floats used, padded to 16 for
  // aligned b128 loads:  p[0..4]=w1[0..4][c0], p[5]=b1[c0],
  //                      p[6..10]=w1[0..4][c0+1], p[11]=b1[c0+1]
  __shared__ float s_w1p[KS * 2 * 16];        // 2 KB

  for (int i = threadIdx.x; i < NT * KS * 32; i += blockDim.x) {
    int lane = i & 31;
    int s    = (i >> 5) & (KS - 1);
    int nt   = i >> 9;
    int koff = (lane >> 4) << 1;
    int n    = nt * 16 + (lane & 15);
    int k0   = 4 * s + koff;
    s_w2[i * 2 + 0] = w2[(k0 + 0) * EMB + n];
    s_w2[i * 2 + 1] = w2[(k0 + 1) * EMB + n];
  }
  for (int i = threadIdx.x; i < KS * 2 * 12; i += blockDim.x) {
    int s    = i / 24;
    int rem  = i % 24;
    int half = rem / 12;
    int q    = rem % 12;
    int c    = 4 * s + 2 * half + (q >= 6 ? 1 : 0);
    int r    = (q >= 6) ? (q - 6) : q;
    s_w1p[(s * 2 + half) * 16 + q] = (r < 5) ? w1[r * HID + c] : b1[c];
  }
  __syncthreads();

  const int lane = threadIdx.x & 31;
  const int m    = lane & 15;                 // row within 16-pillar tile
  const int hsel = lane >> 4;                 // which 16-lane half

  // per-lane epilogue constants (depend only on n = nt*16 + m)
  float g_r[NT], be_r[NT], b2_r[NT];
#pragma unroll
  for (int nt = 0; nt < NT; ++nt) {
    g_r[nt]  = gamma[nt * 16 + m];
    be_r[nt] = beta [nt * 16 + m];
    b2_r[nt] = b2   [nt * 16 + m];
  }

  const int waveId = (blockIdx.x * blockDim.x + threadIdx.x) >> 5;
  const int nWaves = (gridDim.x * blockDim.x) >> 5;
  const int nTiles = BK / 16;                 // 40000

  for (int tile = waveId; tile < nTiles; tile += nWaves) {
    const int g = tile * 16 + m;              // global pillar (both halves load same)
    float cnt = acc[4 * BK + g];
    float inv = 1.0f / fmaxf(cnt, 1.0f);
    float f0 = acc[0 * BK + g] * inv;
    float f1 = acc[1 * BK + g] * inv;
    float f2 = acc[2 * BK + g] * inv;
    float f3 = acc[3 * BK + g] * inv;
    float f4 = cnt;

    // accumulators initialized with b2 bias (bias depends only on n)
    v8f c[NT];
#pragma unroll
    for (int nt = 0; nt < NT; ++nt) {
      float bb = b2_r[nt];
      v8f cc = {bb, bb, bb, bb, bb, bb, bb, bb};
      c[nt] = cc;
    }

    // --- fused GEMM1 (VALU, K=5) + GEMM2 (WMMA fp32) ---
    // Rolled K-loop: 12 live B VGPRs, A pair recomputed per step.
#pragma unroll 1
    for (int s = 0; s < KS; ++s) {
      const float* p = &s_w1p[(s * 2 + hsel) * 16];
      float h0 = fmaf(f0, p[0], fmaf(f1, p[1], fmaf(f2, p[2],
                 fmaf(f3, p[3], fmaf(f4, p[4], p[5])))));
      float h1 = fmaf(f0, p[6], fmaf(f1, p[7], fmaf(f2, p[8],
                 fmaf(f3, p[9], fmaf(f4, p[10], p[11])))));
      v2f av;
      av.x = fmaxf(h0, 0.0f);
      av.y = fmaxf(h1, 0.0f);

      const float* bbase = &s_w2[(s * 32 + lane) * 2];
#pragma unroll
      for (int nt = 0; nt < NT; ++nt) {
        v2f bv = *(const v2f*)&bbase[nt * (KS * 32 * 2)];  // imm offset nt*4KB
        c[nt] = __builtin_amdgcn_wmma_f32_16x16x4_f32(
            false, av, false, bv, (short)0, c[nt], false, false);
      }
    }

    // --- LayerNorm over 96: row j+8*hsel lives in C-vgpr j across a half ---
    float mu[8], rstd[8];
#pragma unroll
    for (int j = 0; j < 8; ++j) {
      float pS = 0.0f;
#pragma unroll
      for (int nt = 0; nt < NT; ++nt) pS += c[nt][j];
      pS += __shfl_xor(pS, 1, 32);
      pS += __shfl_xor(pS, 2, 32);
      pS += __shfl_xor(pS, 4, 32);
      pS += __shfl_xor(pS, 8, 32);           // reduce within each 16-lane half
      mu[j] = pS * (1.0f / 96.0f);
    }
#pragma unroll
    for (int j = 0; j < 8; ++j) {
      float v = 0.0f;
#pragma unroll
      for (int nt = 0; nt < NT; ++nt) {
        float d = c[nt][j] - mu[j];
        v = fmaf(d, d, v);
      }
      v += __shfl_xor(v, 1, 32);
      v += __shfl_xor(v, 2, 32);
      v += __shfl_xor(v, 4, 32);
      v += __shfl_xor(v, 8, 32);
      rstd[j] = rsqrtf(v * (1.0f / 96.0f) + LNEPS);
    }

    // --- normalize + affine + store ---
#pragma unroll
    for (int j = 0; j < 8; ++j) {
      int mrow = j + 8 * hsel;
      float* orow = out + (size_t)(tile * 16 + mrow) * EMB;
#pragma unroll
      for (int nt = 0; nt < NT; ++nt) {
        float val = fmaf((c[nt][j] - mu[j]) * rstd[j], g_r[nt], be_r[nt]);
        orow[nt * 16 + m] = val;
      }
    }
  }
}

// --------------------------------------------------------------------------
extern "C" void kernel_launch(void* const* d_in, const int* in_sizes, int n_in,
                              void* d_out, int out_size, void* d_ws, size_t ws_size,
                              hipStream_t stream) {
  const float* points = (const float*)d_in[0];
  const float* w1     = (const float*)d_in[1];
  const float* b1     = (const float*)d_in[2];
  const float* w2     = (const float*)d_in[3];
  const float* b2     = (const float*)d_in[4];
  const float* gamma  = (const float*)d_in[5];
  const float* beta   = (const float*)d_in[6];
  float* out = (float*)d_out;
  float* acc = (float*)d_ws;                 // 5 * BK floats = 12.8 MB scratch

  pillar_zero<<<512, 256, 0, stream>>>(acc, 5 * BK);

  const int total = BATCH * NPTS;            // 3.2M points
  pillar_scatter<<<(total + 255) / 256, 256, 0, stream>>>(
      (const float4*)points, acc, total);

  pillar_mlp_ln<<<800, 256, 0, stream>>>(acc, w1, b1, w2, b2, gamma, beta, out);
}